// Conv1dMultiscaleLocalization_77627238908099
// MI455X (gfx1250) — compile-verified
//
#include <hip/hip_runtime.h>
#include <hip/hip_bf16.h>
#include <math.h>

// ---------------- problem constants ----------------
#define BB   8
#define HH   768
#define WW   768
#define NS   7          // number of scales
#define MAXK 65         // padded kernel length
#define HALF 32         // MAXK/2
#define KDIM 80         // 16 outputs + 64 halo -> contraction length
#define KST  20         // KDIM / 4 k-steps of V_WMMA_F32_16X16X4_F32
#define WPAD 96         // padded per-scale weight row in LDS: 16 zeros | 65 taps | 15 zeros
#define TILES_X (WW/16) // 48
#define TILES_Y (HH/16) // 48
#define TILES_PER_IMG (TILES_X*TILES_Y)
#define MIN_DIST 5

typedef float v2f __attribute__((ext_vector_type(2)));
typedef float v8f __attribute__((ext_vector_type(8)));

// ---------------------------------------------------------------------------
// fp32 WMMA tile worker. One wave = one 16x16 output tile.
//   vertical  (C): acc[s] += Wband(16x80) x Cwin(80x16)
//   horizontal(S): acc[s] += Swin(16x80) x Wband(80x16)
// f32 16x16x4 fragment striping (CDNA5 ISA):
//   A: VGPR0 = {M=lane%16, K=0 | K=2}, VGPR1 = {K=1 | K=3}
//   B: VGPR0 = {K=0 | K=2, N=lane%16}, VGPR1 = {K=1 | K=3}
// -> the banded-weight fragment is the same v2f for both roles:
//    w[s, 4*kt + (lane<16?0:2) + {0,1} - lane%16], fetched from the
//    zero-padded LDS row so no bounds selects are needed.
// ---------------------------------------------------------------------------
template<bool INTERIOR>
__device__ __forceinline__ void conv_tile(const float* __restrict__ Cb,
                                          const float* __restrict__ Sb,
                                          const float* __restrict__ wlds,
                                          int y0, int x0, int p, int kkbase,
                                          v8f (&acc)[NS])
{
    const int sy = y0 + p;                       // always in [0,HH)
    const int cx = x0 + p;                       // always in [0,WW)
    const float* __restrict__ srow = Sb + (size_t)sy * WW + (x0 - HALF);
    const float* __restrict__ ccol = Cb + (size_t)(y0 - HALF) * WW + cx;
    const float* __restrict__ wbase = wlds + (16 + kkbase - p);   // lane-constant

    #pragma unroll 4
    for (int kt = 0; kt < KST; ++kt) {
        const int tbase = 4 * kt + kkbase;

        v2f aS, bC;
        if (INTERIOR) {
            aS.x = srow[tbase];
            aS.y = srow[tbase + 1];
            bC.x = ccol[(size_t)tbase * WW];
            bC.y = ccol[(size_t)(tbase + 1) * WW];
        } else {
            const int sx0 = x0 - HALF + tbase;
            const int cy0 = y0 - HALF + tbase;
            aS.x = ((unsigned)sx0       < (unsigned)WW) ? srow[tbase]                 : 0.0f;
            aS.y = ((unsigned)(sx0 + 1) < (unsigned)WW) ? srow[tbase + 1]             : 0.0f;
            bC.x = ((unsigned)cy0       < (unsigned)HH) ? ccol[(size_t)tbase * WW]    : 0.0f;
            bC.y = ((unsigned)(cy0 + 1) < (unsigned)HH) ? ccol[(size_t)(tbase+1)*WW]  : 0.0f;
        }

        // gather all 7 weight fragments (padded LDS: no bounds checks)
        v2f wf[NS];
        #pragma unroll
        for (int s = 0; s < NS; ++s) {
            wf[s].x = wbase[s * WPAD + 4 * kt];
            wf[s].y = wbase[s * WPAD + 4 * kt + 1];
        }
        // 7 independent vertical WMMAs, then 7 independent horizontal ones:
        // accumulation chains on any acc[s] are >= 7 instructions apart.
        #pragma unroll
        for (int s = 0; s < NS; ++s)
            acc[s] = __builtin_amdgcn_wmma_f32_16x16x4_f32(
                false, wf[s], false, bC, (short)0, acc[s], false, false);
        #pragma unroll
        for (int s = 0; s < NS; ++s)
            acc[s] = __builtin_amdgcn_wmma_f32_16x16x4_f32(
                false, aS, false, wf[s], (short)0, acc[s], false, false);
    }
}

__global__ __launch_bounds__(256)
void conv_wmma_kernel(const float* __restrict__ Cin,
                      const float* __restrict__ Sin,
                      const float* __restrict__ wk,   // kernel_cos flat [7*65]
                      float* __restrict__ conv_out)
{
    // zero-padded weights: wlds[s*96 + 16 + j] = wk[s*65 + j], zeros elsewhere
    __shared__ float wlds[NS * WPAD];
    for (int i = threadIdx.x; i < NS * WPAD; i += 256) {
        const int s = i / WPAD;
        const int j = (i % WPAD) - 16;
        wlds[i] = ((unsigned)j < (unsigned)MAXK) ? wk[s * MAXK + j] : 0.0f;
    }
    __syncthreads();

    const int lane = threadIdx.x & 31;
    const int wave = threadIdx.x >> 5;
    const int tile = blockIdx.x * 8 + wave;

    const int b  = tile / TILES_PER_IMG;
    const int t  = tile % TILES_PER_IMG;
    const int ty = t / TILES_X;
    const int tx = t % TILES_X;
    const int y0 = ty * 16;
    const int x0 = tx * 16;

    const float* __restrict__ Cb = Cin + (size_t)b * HH * WW;
    const float* __restrict__ Sb = Sin + (size_t)b * HH * WW;

    const int p      = lane & 15;
    const int kkbase = (lane < 16) ? 0 : 2;

    v8f acc[NS];
    #pragma unroll
    for (int s = 0; s < NS; ++s) acc[s] = (v8f)(0.0f);

    // uniform per-wave branch: 44x44 of 48x48 tiles take the unchecked path
    const bool interior = (tx >= 2) & (tx <= TILES_X - 3) & (ty >= 2) & (ty <= TILES_Y - 3);
    if (interior) conv_tile<true >(Cb, Sb, wlds, y0, x0, p, kkbase, acc);
    else          conv_tile<false>(Cb, Sb, wlds, y0, x0, p, kkbase, acc);

    // max over the 7 scales
    v8f m = acc[0];
    #pragma unroll
    for (int s = 1; s < NS; ++s)
        #pragma unroll
        for (int i = 0; i < 8; ++i)
            m[i] = fmaxf(m[i], acc[s][i]);

    // D layout store: VGPR i -> row y0 + i + (lane<16?0:8), col x0 + p
    float* __restrict__ outp = conv_out + (size_t)b * HH * WW;
    const int rbase = y0 + ((lane < 16) ? 0 : 8);
    #pragma unroll
    for (int i = 0; i < 8; ++i)
        outp[(size_t)(rbase + i) * WW + x0 + p] = m[i];
}

// ---------------------------------------------------------------------------
// Pass 2a: horizontal 11-wide max of pos = relu(conv_resp) -> ws
// ---------------------------------------------------------------------------
__global__ __launch_bounds__(256)
void hmax_kernel(const float* __restrict__ resp, float* __restrict__ hmax)
{
    const int idx = blockIdx.x * 256 + threadIdx.x;
    if (idx >= BB * HH * WW) return;
    const int x = idx % WW;
    const int rowbase = idx - x;               // row boundaries == image boundaries
    const int xl = (x - MIN_DIST) > 0 ? (x - MIN_DIST) : 0;
    const int xr = (x + MIN_DIST) < (WW - 1) ? (x + MIN_DIST) : (WW - 1);
    float m = -INFINITY;
    for (int xx = xl; xx <= xr; ++xx)
        m = fmaxf(m, fmaxf(resp[rowbase + xx], 0.0f));
    hmax[idx] = m;
}

// ---------------------------------------------------------------------------
// Pass 2b: vertical 11-wide max of ws -> pooled; peak mask as 0.0/1.0 floats
// ---------------------------------------------------------------------------
__global__ __launch_bounds__(256)
void vmask_kernel(const float* __restrict__ resp, const float* __restrict__ hmax,
                  float* __restrict__ mask)
{
    const int idx = blockIdx.x * 256 + threadIdx.x;
    if (idx >= BB * HH * WW) return;
    const int x = idx % WW;
    const int y = (idx / WW) % HH;
    const int b = idx / (WW * HH);
    const int yl = (y - MIN_DIST) > 0 ? (y - MIN_DIST) : 0;
    const int yr = (y + MIN_DIST) < (HH - 1) ? (y + MIN_DIST) : (HH - 1);
    const float* __restrict__ hb = hmax + (size_t)b * HH * WW;
    float pooled = -INFINITY;
    for (int yy = yl; yy <= yr; ++yy)
        pooled = fmaxf(pooled, hb[(size_t)yy * WW + x]);
    const float pos = fmaxf(resp[idx], 0.0f);
    mask[idx] = (pos == pooled && pos > 0.5f) ? 1.0f : 0.0f;
}

// ---------------------------------------------------------------------------
extern "C" void kernel_launch(void* const* d_in, const int* in_sizes, int n_in,
                              void* d_out, int out_size, void* d_ws, size_t ws_size,
                              hipStream_t stream)
{
    (void)in_sizes; (void)n_in; (void)out_size; (void)ws_size;

    const float* C    = (const float*)d_in[0];
    const float* S    = (const float*)d_in[1];
    const float* kcos = (const float*)d_in[2];      // [7,1,65,1] == flat [7*65]

    const size_t npix = (size_t)BB * HH * WW;
    float* conv_out = (float*)d_out;                // output 0: conv_resp
    float* mask_out = (float*)d_out + npix;         // output 1: mask (0/1 floats)
    float* hmax     = (float*)d_ws;                 // needs npix*4 = 18.9 MB scratch

    const int ntiles = BB * TILES_PER_IMG;          // 18432, 8 tiles/block
    conv_wmma_kernel<<<ntiles / 8, 256, 0, stream>>>(C, S, kcos, conv_out);

    const int nblk = (int)((npix + 255) / 256);
    hmax_kernel <<<nblk, 256, 0, stream>>>(conv_out, hmax);
    vmask_kernel<<<nblk, 256, 0, stream>>>(conv_out, hmax, mask_out);
}